// GraphGATClassifier_28260884808097
// MI455X (gfx1250) — compile-verified
//
#include <hip/hip_runtime.h>
#include <hip/hip_bf16.h>

#define N_NODES  50000
#define N_EDGES  800000
#define DFEAT    128
#define N_GRAPHS 64
#define N_CLASSES 2
#define SLOPE    0.2f

typedef float v2f __attribute__((ext_vector_type(2)));
typedef float v8f __attribute__((ext_vector_type(8)));

// ---------------------------------------------------------------------------
// Utility: zero an int buffer
// ---------------------------------------------------------------------------
__global__ void zero_i32(int* p, int n) {
    int i = blockIdx.x * blockDim.x + threadIdx.x;
    if (i < n) p[i] = 0;
}

// ---------------------------------------------------------------------------
// CSR build: in-degree count -> exclusive scan -> scatter fill
// ---------------------------------------------------------------------------
__global__ void count_deg(const int* __restrict__ dst, int* __restrict__ deg) {
    int e = blockIdx.x * blockDim.x + threadIdx.x;
    if (e < N_EDGES) atomicAdd(&deg[dst[e]], 1);
}

__global__ __launch_bounds__(1024) void scan_deg(const int* __restrict__ deg,
                                                 int* __restrict__ off,
                                                 int* __restrict__ cur, int n) {
    __shared__ int buf[1024];
    __shared__ int base_s;
    int t = threadIdx.x;
    if (t == 0) base_s = 0;
    __syncthreads();
    for (int start = 0; start < n; start += 1024) {
        int i = start + t;
        int v = (i < n) ? deg[i] : 0;
        buf[t] = v;
        __syncthreads();
        // Hillis-Steele inclusive scan
        for (int s = 1; s < 1024; s <<= 1) {
            int add = (t >= s) ? buf[t - s] : 0;
            __syncthreads();
            buf[t] += add;
            __syncthreads();
        }
        int excl = base_s + buf[t] - v;
        if (i < n) { off[i] = excl; cur[i] = excl; }
        __syncthreads();           // everyone read base_s before update
        if (t == 0) base_s += buf[1023];
        __syncthreads();
    }
    if (t == 0) off[n] = base_s;
}

__global__ void fill_csr(const int* __restrict__ src, const int* __restrict__ dst,
                         int* __restrict__ cur, int* __restrict__ csr_src) {
    int e = blockIdx.x * blockDim.x + threadIdx.x;
    if (e < N_EDGES) {
        int d = dst[e];
        int pos = atomicAdd(&cur[d], 1);
        csr_src[pos] = src[e];
    }
}

// ---------------------------------------------------------------------------
// GEMM:  C[M x 128] = A[M x 128] * B[128 x 128]  (row-major, f32 WMMA)
// block = 256 threads = 8 waves; block owns 16 rows, wave w owns N-tile w.
// ---------------------------------------------------------------------------
__global__ __launch_bounds__(256) void gemm_wmma_f32(const float* __restrict__ A,
                                                     const float* __restrict__ B,
                                                     float* __restrict__ C) {
    const int wave = threadIdx.x >> 5;          // 0..7 -> N tile
    const int lane = threadIdx.x & 31;
    const int tileM = blockIdx.x;               // 0..3124
    const int row = lane & 15;
    const int kq  = (lane >> 4) << 1;           // 0 or 2 (K sub-pair)
    const float* arow = A + (size_t)(tileM * 16 + row) * DFEAT;
    const int col = (lane & 15) + wave * 16;    // global N column
    v8f c = {};
    #pragma unroll 4
    for (int k = 0; k < DFEAT; k += 4) {
        v2f a, b;
        a.x = arow[k + kq];
        a.y = arow[k + kq + 1];
        b.x = B[(size_t)(k + kq)     * DFEAT + col];
        b.y = B[(size_t)(k + kq + 1) * DFEAT + col];
        c = __builtin_amdgcn_wmma_f32_16x16x4_f32(false, a, false, b,
                                                  (short)0, c, false, false);
    }
    const int r0 = (lane >> 4) << 3;            // 0 or 8
    float* crow = C + (size_t)(tileM * 16) * DFEAT + col;
    #pragma unroll
    for (int r = 0; r < 8; ++r)
        crow[(size_t)(r0 + r) * DFEAT] = c[r];
}

// ---------------------------------------------------------------------------
// el/er: per-node dot products of Wh row with al / ar. One wave per node.
// ---------------------------------------------------------------------------
__global__ __launch_bounds__(256) void eler_kernel(const float* __restrict__ Wh,
                                                   const float* __restrict__ al,
                                                   const float* __restrict__ ar,
                                                   float* __restrict__ el,
                                                   float* __restrict__ er) {
    int node = (blockIdx.x * blockDim.x + threadIdx.x) >> 5;
    int lane = threadIdx.x & 31;
    if (node >= N_NODES) return;
    const float4 v  = ((const float4*)(Wh + (size_t)node * DFEAT))[lane];
    const float4 a4 = ((const float4*)al)[lane];
    const float4 r4 = ((const float4*)ar)[lane];
    float dl = v.x * a4.x + v.y * a4.y + v.z * a4.z + v.w * a4.w;
    float dr = v.x * r4.x + v.y * r4.y + v.z * r4.z + v.w * r4.w;
    #pragma unroll
    for (int m = 16; m >= 1; m >>= 1) {
        dl += __shfl_xor(dl, m, 32);
        dr += __shfl_xor(dr, m, 32);
    }
    if (lane == 0) { el[node] = dl; er[node] = dr; }
}

// ---------------------------------------------------------------------------
// GAT aggregation: one block (128 threads = 1 feature each) per dst node.
// Fused segment-softmax (max pass + single exp/accumulate pass), bias, ELU.
// No atomics; deterministic per CSR ordering.
// ---------------------------------------------------------------------------
__global__ __launch_bounds__(128) void gat_aggregate(const int* __restrict__ off,
                                                     const int* __restrict__ csr_src,
                                                     const float* __restrict__ el,
                                                     const float* __restrict__ er,
                                                     const float* __restrict__ Wh,
                                                     const float* __restrict__ bias,
                                                     float* __restrict__ hout) {
    const int n = blockIdx.x;
    const int t = threadIdx.x;                  // feature index
    const int beg = off[n], end = off[n + 1];
    float val;
    if (beg == end) {
        val = bias[t];                          // zero in-degree: bias only
    } else {
        const float ern = er[n];
        // pass 1: max score (redundantly computed by all threads; deg avg ~16)
        float m = -3.4e38f;
        for (int j = beg; j < end; ++j) {
            float sc = el[csr_src[j]] + ern;
            sc = (sc > 0.f) ? sc : SLOPE * sc;
            m = fmaxf(m, sc);
        }
        // pass 2: exp-sum + weighted feature accumulation (un-normalized)
        float acc = 0.f, den = 0.f;
        for (int j = beg; j < end; ++j) {
            int s = csr_src[j];
            float sc = el[s] + ern;
            sc = (sc > 0.f) ? sc : SLOPE * sc;
            float ex = expf(sc - m);
            den += ex;
            acc += ex * Wh[(size_t)s * DFEAT + t];
        }
        val = acc / den + bias[t];
    }
    // ELU
    val = (val > 0.f) ? val : expm1f(val);
    hout[(size_t)n * DFEAT + t] = val;
}

// ---------------------------------------------------------------------------
// Readout: graph_ids is sorted -> per-graph contiguous node range via binary
// search. One block per graph (128 threads = features). Deterministic.
// ---------------------------------------------------------------------------
__device__ __forceinline__ int lower_bound_i(const int* a, int n, int key) {
    int lo = 0, hi = n;
    while (lo < hi) {
        int mid = (lo + hi) >> 1;
        if (a[mid] < key) lo = mid + 1; else hi = mid;
    }
    return lo;
}

__global__ __launch_bounds__(128) void readout_kernel(const float* __restrict__ h3,
                                                      const int* __restrict__ gids,
                                                      const float* __restrict__ Wc,
                                                      const float* __restrict__ bc,
                                                      float* __restrict__ out) {
    const int g = blockIdx.x;
    const int t = threadIdx.x;                  // feature
    const int start = lower_bound_i(gids, N_NODES, g);
    const int stop  = lower_bound_i(gids, N_NODES, g + 1);
    float s = 0.f;
    for (int n = start; n < stop; ++n) s += h3[(size_t)n * DFEAT + t];
    float cnt = (float)((stop - start) > 0 ? (stop - start) : 1);
    float mean = s / cnt;

    __shared__ float red[128];
    #pragma unroll
    for (int c = 0; c < N_CLASSES; ++c) {
        red[t] = mean * Wc[t * N_CLASSES + c];
        __syncthreads();
        for (int sft = 64; sft >= 1; sft >>= 1) {
            if (t < sft) red[t] += red[t + sft];
            __syncthreads();
        }
        if (t == 0) out[g * N_CLASSES + c] = red[0] + bc[c];
        __syncthreads();
    }
}

// ---------------------------------------------------------------------------
// Host launcher
// ---------------------------------------------------------------------------
extern "C" void kernel_launch(void* const* d_in, const int* in_sizes, int n_in,
                              void* d_out, int out_size, void* d_ws, size_t ws_size,
                              hipStream_t stream) {
    const float* x    = (const float*)d_in[0];
    const int*   src  = (const int*)d_in[1];
    const int*   dst  = (const int*)d_in[2];
    const int*   gids = (const int*)d_in[3];
    const float* W[3]  = {(const float*)d_in[4],  (const float*)d_in[8],  (const float*)d_in[12]};
    const float* al[3] = {(const float*)d_in[5],  (const float*)d_in[9],  (const float*)d_in[13]};
    const float* ar[3] = {(const float*)d_in[6],  (const float*)d_in[10], (const float*)d_in[14]};
    const float* bl[3] = {(const float*)d_in[7],  (const float*)d_in[11], (const float*)d_in[15]};
    const float* Wc = (const float*)d_in[16];
    const float* bc = (const float*)d_in[17];
    float* out = (float*)d_out;

    // ---- workspace carve-up (256B aligned) ----
    char* w = (char*)d_ws;
    auto take = [&](size_t bytes) {
        void* p = (void*)w;
        w += (bytes + 255) & ~size_t(255);
        return p;
    };
    float* Wh  = (float*)take((size_t)N_NODES * DFEAT * 4);
    float* hA  = (float*)take((size_t)N_NODES * DFEAT * 4);
    float* hB  = (float*)take((size_t)N_NODES * DFEAT * 4);
    float* el  = (float*)take((size_t)N_NODES * 4);
    float* er  = (float*)take((size_t)N_NODES * 4);
    int* deg   = (int*)take((size_t)N_NODES * 4);
    int* off   = (int*)take((size_t)(N_NODES + 1) * 4);
    int* cur   = (int*)take((size_t)N_NODES * 4);
    int* csrc  = (int*)take((size_t)N_EDGES * 4);

    // ---- build CSR (dst-grouped) once; reused by all 3 layers ----
    zero_i32<<<(N_NODES + 255) / 256, 256, 0, stream>>>(deg, N_NODES);
    count_deg<<<(N_EDGES + 255) / 256, 256, 0, stream>>>(dst, deg);
    scan_deg<<<1, 1024, 0, stream>>>(deg, off, cur, N_NODES);
    fill_csr<<<(N_EDGES + 255) / 256, 256, 0, stream>>>(src, dst, cur, csrc);

    // ---- 3 GAT layers ----
    const float* hin[3] = {x, hA, hB};
    float* hout[3]      = {hA, hB, hA};
    for (int l = 0; l < 3; ++l) {
        gemm_wmma_f32<<<N_NODES / 16, 256, 0, stream>>>(hin[l], W[l], Wh);
        eler_kernel<<<(N_NODES * 32 + 255) / 256, 256, 0, stream>>>(Wh, al[l], ar[l], el, er);
        gat_aggregate<<<N_NODES, 128, 0, stream>>>(off, csrc, el, er, Wh, bl[l], hout[l]);
    }

    // ---- per-graph mean + classifier ----
    readout_kernel<<<N_GRAPHS, 128, 0, stream>>>(hA, gids, Wc, bc, out);
}